// model_4v4_22067541967342
// MI455X (gfx1250) — compile-verified
//
#include <hip/hip_runtime.h>
#include <math.h>

#define D 128
#define H3 384
#define HB 256
#define NEG_SLOPE 0.01f
#define BN_EPS 1e-5f

typedef __attribute__((ext_vector_type(2))) float v2f;
typedef __attribute__((ext_vector_type(8))) float v8f;

// ---------------------------------------------------------------- utilities

__global__ void fill_kernel(float* __restrict__ p, long n, float v) {
    long i = (long)blockIdx.x * blockDim.x + threadIdx.x;
    if (i < n) p[i] = v;
}

// float atomic max via sign-split integer atomics (works with -inf init)
__device__ __forceinline__ void atomicMaxF32(float* addr, float val) {
    if (val >= 0.0f) atomicMax((int*)addr, __float_as_int(val));
    else             atomicMin((unsigned int*)addr, __float_as_uint(val));
}

// ------------------------------------------------------- edge aggregation
// one wave32 per edge; lane l moves x[src, 4l..4l+3] (b128 load per lane)
__global__ __launch_bounds__(256) void edge_aggregate(
    const int* __restrict__ src, const int* __restrict__ dst,
    const float* __restrict__ x,
    float* __restrict__ asum, float* __restrict__ amax, int E) {
    int lane = threadIdx.x & 31;
    int wave = threadIdx.x >> 5;
    int e = blockIdx.x * 8 + wave;
    if (e >= E) return;
    int s = src[e];
    int d = dst[e];
    const float4* xr = (const float4*)(x + (size_t)s * D);
    float4 v = xr[lane];
    float* ps = asum + (size_t)d * D + lane * 4;
    float* pm = amax + (size_t)d * D + lane * 4;
    atomicAdd(ps + 0, v.x); atomicAdd(ps + 1, v.y);
    atomicAdd(ps + 2, v.z); atomicAdd(ps + 3, v.w);
    atomicMaxF32(pm + 0, v.x); atomicMaxF32(pm + 1, v.y);
    atomicMaxF32(pm + 2, v.z); atomicMaxF32(pm + 3, v.w);
}

// ------------------------------------------------------------ WMMA GEMM
// C[N, DOUT] = act( Aload(N, DIN) @ B + bias )
// AMODE 0: plain A.  AMODE 1: concat(sum, max-with--inf->0), two branchless
// K-loops.  AMODE 2: alpha[k]*A+beta[k] (BN folded into A operand).
// BT: B[k,col] = B[col*LDB + k] (X @ W.T), contiguous -> b64 loads.
// Block = 256 thr = 8 waves = 2 row-tiles x 4 col-groups; each wave owns a
// 16x32 output strip (2 accumulators -> 2 wmma per A fragment).
template<bool BT, int LDB>
__device__ __forceinline__ void wmma_k4(const float* __restrict__ Bb, int ka,
                                        int col0, int col1, float a0, float a1,
                                        v8f& acc0, v8f& acc1) {
    v2f av; av.x = a0; av.y = a1;
    v2f bv0, bv1;
    if (BT) {
        float2 b0 = *(const float2*)(Bb + (size_t)col0 * LDB + ka);
        float2 b1 = *(const float2*)(Bb + (size_t)col1 * LDB + ka);
        bv0.x = b0.x; bv0.y = b0.y; bv1.x = b1.x; bv1.y = b1.y;
    } else {
        bv0.x = Bb[(size_t)ka * LDB + col0];
        bv0.y = Bb[(size_t)(ka + 1) * LDB + col0];
        bv1.x = Bb[(size_t)ka * LDB + col1];
        bv1.y = Bb[(size_t)(ka + 1) * LDB + col1];
    }
    acc0 = __builtin_amdgcn_wmma_f32_16x16x4_f32(false, av, false, bv0, (short)0, acc0, false, false);
    acc1 = __builtin_amdgcn_wmma_f32_16x16x4_f32(false, av, false, bv1, (short)0, acc1, false, false);
}

template<int AMODE, bool BT, bool LEAKY, int DIN, int LDA, int LDB, int DOUT>
__global__ __launch_bounds__(256) void wmma_gemm(
    const float* __restrict__ A, const float* __restrict__ Aux0,
    const float* __restrict__ Aux1,
    const float* __restrict__ B, const float* __restrict__ bias,
    float* __restrict__ C) {
    const int lane = threadIdx.x & 31;
    const int wave = threadIdx.x >> 5;
    const int wrow = wave >> 2;                       // 0..1
    const int wcol = wave & 3;                        // 0..3
    const int l16  = lane & 15;
    const int half = lane >> 4;                       // A/B frag: v0=K,K+2 ; v1=K+1,K+3
    const long rowA = (long)blockIdx.x * 32 + wrow * 16 + l16;
    const int  col0 = blockIdx.y * 128 + wcol * 32 + l16;
    const int  col1 = col0 + 16;

    const float* __restrict__ Ar = A + rowA * LDA;
    v8f acc0 = {}, acc1 = {};

    if (AMODE == 1) {
        const float* __restrict__ Am = Aux0 + rowA * LDA;   // max half
        const float* __restrict__ B2 = B + (BT ? (size_t)LDA : (size_t)LDA * LDB);
#pragma unroll 8
        for (int k0 = 0; k0 < LDA; k0 += 4) {
            const int ka = k0 + half * 2;
            float2 a = *(const float2*)(Ar + ka);
            wmma_k4<BT, LDB>(B, ka, col0, col1, a.x, a.y, acc0, acc1);
        }
#pragma unroll 8
        for (int k0 = 0; k0 < LDA; k0 += 4) {
            const int ka = k0 + half * 2;
            float2 a = *(const float2*)(Am + ka);
            float a0 = (a.x < -3.0e38f) ? 0.0f : a.x;    // isneginf -> 0
            float a1 = (a.y < -3.0e38f) ? 0.0f : a.y;
            wmma_k4<BT, LDB>(B2, ka, col0, col1, a0, a1, acc0, acc1);
        }
    } else {
#pragma unroll 8
        for (int k0 = 0; k0 < DIN; k0 += 4) {
            const int ka = k0 + half * 2;
            float2 a = *(const float2*)(Ar + ka);
            float a0 = a.x, a1 = a.y;
            if (AMODE == 2) {
                float2 al = *(const float2*)(Aux0 + ka);
                float2 be = *(const float2*)(Aux1 + ka);
                a0 = al.x * a0 + be.x;
                a1 = al.y * a1 + be.y;
            }
            wmma_k4<BT, LDB>(B, ka, col0, col1, a0, a1, acc0, acc1);
        }
    }

    const float bs0 = bias[col0];
    const float bs1 = bias[col1];
    const size_t rbase = (size_t)blockIdx.x * 32 + wrow * 16 + half * 8;
#pragma unroll
    for (int r = 0; r < 8; ++r) {                     // C VGPR r: M = r (+8 hi half)
        float v0 = acc0[r] + bs0;
        float v1 = acc1[r] + bs1;
        if (LEAKY) {
            v0 = (v0 >= 0.0f) ? v0 : NEG_SLOPE * v0;
            v1 = (v1 >= 0.0f) ? v1 : NEG_SLOPE * v1;
        }
        C[(rbase + r) * DOUT + col0] = v0;
        C[(rbase + r) * DOUT + col1] = v1;
    }
}

// --------------------------------------------------------------- GRU gates
// float4 per thread: i in [0, N*D/4)
__global__ void gru_gate(const float4* __restrict__ gi, const float4* __restrict__ gh,
                         const float4* __restrict__ x, float4* __restrict__ h, long n4) {
    long i = (long)blockIdx.x * blockDim.x + threadIdx.x;
    if (i >= n4) return;
    long row = i >> 5;                 // / (128/4)
    int  c4  = (int)(i & 31);          // float4 index within row
    size_t b = (size_t)row * (H3 / 4) + c4;
    float4 ir = gi[b], iz = gi[b + 32], in_ = gi[b + 64];
    float4 hr = gh[b], hz = gh[b + 32], hn  = gh[b + 64];
    float4 xv = x[i];
    float4 o;
    {
        float r  = 1.0f / (1.0f + __expf(-(ir.x + hr.x)));
        float z  = 1.0f / (1.0f + __expf(-(iz.x + hz.x)));
        o.x = (1.0f - z) * tanhf(in_.x + r * hn.x) + z * xv.x;
    }
    {
        float r  = 1.0f / (1.0f + __expf(-(ir.y + hr.y)));
        float z  = 1.0f / (1.0f + __expf(-(iz.y + hz.y)));
        o.y = (1.0f - z) * tanhf(in_.y + r * hn.y) + z * xv.y;
    }
    {
        float r  = 1.0f / (1.0f + __expf(-(ir.z + hr.z)));
        float z  = 1.0f / (1.0f + __expf(-(iz.z + hz.z)));
        o.z = (1.0f - z) * tanhf(in_.z + r * hn.z) + z * xv.z;
    }
    {
        float r  = 1.0f / (1.0f + __expf(-(ir.w + hr.w)));
        float z  = 1.0f / (1.0f + __expf(-(iz.w + hz.w)));
        o.w = (1.0f - z) * tanhf(in_.w + r * hn.w) + z * xv.w;
    }
    h[i] = o;
}

// ---------------------------------------------------------------- batchnorm
// per-column sum / sumsq (C power of two: 256 or 128)
__global__ __launch_bounds__(256) void col_stats(const float* __restrict__ Y, int n, int C,
                                                 float* __restrict__ sums) {
    int tid = threadIdx.x;
    int col = tid & (C - 1);
    int rpb = 256 / C;
    int r0  = blockIdx.x * rpb + (tid / C);
    int stride = gridDim.x * rpb;
    float s = 0.0f, ss = 0.0f;
    for (int r = r0; r < n; r += stride) {
        float v = Y[(size_t)r * C + col];
        s += v; ss += v * v;
    }
    atomicAdd(&sums[col], s);
    atomicAdd(&sums[C + col], ss);
}

__global__ void bn_affine(const float* __restrict__ sums, const float* __restrict__ g,
                          const float* __restrict__ be, float* __restrict__ alpha,
                          float* __restrict__ beta, int C, float invN) {
    int c = blockIdx.x * blockDim.x + threadIdx.x;
    if (c >= C) return;
    float m = sums[c] * invN;
    float v = sums[C + c] * invN - m * m;   // biased var, like torch/jnp
    float a = g[c] * rsqrtf(v + BN_EPS);
    alpha[c] = a;
    beta[c]  = be[c] - m * a;
}

// float4 per thread; C = 128 -> 32 float4 per row
__global__ void bn_apply(const float4* __restrict__ Y, const float4* __restrict__ alpha,
                         const float4* __restrict__ beta, float4* __restrict__ out, long n4) {
    long i = (long)blockIdx.x * blockDim.x + threadIdx.x;
    if (i >= n4) return;
    int c4 = (int)(i & 31);
    float4 y = Y[i], a = alpha[c4], b = beta[c4];
    float4 o;
    o.x = a.x * y.x + b.x; o.y = a.y * y.y + b.y;
    o.z = a.z * y.z + b.z; o.w = a.w * y.w + b.w;
    out[i] = o;
}

// ------------------------------------------------------------------ driver
extern "C" void kernel_launch(void* const* d_in, const int* in_sizes, int n_in,
                              void* d_out, int out_size, void* d_ws, size_t ws_size,
                              hipStream_t stream) {
    const float* x       = (const float*)d_in[0];
    const int*   edges   = (const int*)d_in[1];
    const float* W_merge = (const float*)d_in[2];
    const float* b_merge = (const float*)d_in[3];
    const float* W_ih    = (const float*)d_in[4];
    const float* W_hh    = (const float*)d_in[5];
    const float* b_ih    = (const float*)d_in[6];
    const float* b_hh    = (const float*)d_in[7];
    const float* W1      = (const float*)d_in[8];
    const float* b1      = (const float*)d_in[9];
    const float* g1      = (const float*)d_in[10];
    const float* be1     = (const float*)d_in[11];
    const float* W2      = (const float*)d_in[12];
    const float* b2      = (const float*)d_in[13];
    const float* g2      = (const float*)d_in[14];
    const float* be2     = (const float*)d_in[15];
    float* out = (float*)d_out;

    const int  Nn = in_sizes[0] / D;      // 100000 (multiple of 32)
    const int  E  = in_sizes[1] / 2;      // 1600000
    const long ND = (long)Nn * D;

    float* ws     = (float*)d_ws;
    float* asum   = ws;                       // N*128
    float* amax   = asum + ND;                // N*128
    float* merged = amax + ND;                // N*128
    float* gi     = merged + ND;              // N*384
    float* gh     = gi + (long)Nn * H3;       // N*384
    float* small_ = gh + (long)Nn * H3;
    float* stats1 = small_;                   // 2*256
    float* alpha1 = stats1 + 2 * HB;
    float* beta1  = alpha1 + HB;
    float* stats2 = beta1 + HB;               // 2*128
    float* alpha2 = stats2 + 2 * D;
    float* beta2  = alpha2 + D;
    // buffer reuse after liveness ends:
    float* hbuf = asum;    // h overwrites agg_sum (dead after merge GEMM)
    float* y1   = gi;      // N*256 fits in N*384 (gi dead after gates)
    float* y2   = amax;    // N*128 (amax dead after merge GEMM)

    // init aggregation buffers
    fill_kernel<<<(int)((ND + 255) / 256), 256, 0, stream>>>(asum, ND, 0.0f);
    fill_kernel<<<(int)((ND + 255) / 256), 256, 0, stream>>>(amax, ND, -INFINITY);
    // gather + scatter (sum, max)
    edge_aggregate<<<(E + 7) / 8, 256, 0, stream>>>(edges, edges + E, x, asum, amax, E);
    // merged = cat(sum, maxfix) @ W_merge + b_merge            (N,256)@(256,128)
    wmma_gemm<1, false, false, 256, 128, 128, 128>
        <<<dim3(Nn / 32, 1), 256, 0, stream>>>(asum, amax, nullptr, W_merge, b_merge, merged);
    // gi = merged @ W_ih.T + b_ih ; gh = x @ W_hh.T + b_hh     (N,128)@(128,384)
    wmma_gemm<0, true, false, 128, 128, 128, 384>
        <<<dim3(Nn / 32, 3), 256, 0, stream>>>(merged, nullptr, nullptr, W_ih, b_ih, gi);
    wmma_gemm<0, true, false, 128, 128, 128, 384>
        <<<dim3(Nn / 32, 3), 256, 0, stream>>>(x, nullptr, nullptr, W_hh, b_hh, gh);
    gru_gate<<<(int)((ND / 4 + 255) / 256), 256, 0, stream>>>(
        (const float4*)gi, (const float4*)gh, (const float4*)x, (float4*)hbuf, ND / 4);
    // y1 = leaky(h @ W1 + b1)                                   (N,128)@(128,256)
    wmma_gemm<0, false, true, 128, 128, 256, 256>
        <<<dim3(Nn / 32, 2), 256, 0, stream>>>(hbuf, nullptr, nullptr, W1, b1, y1);
    // BN1 stats -> affine, folded into next GEMM's A operand
    fill_kernel<<<2, 256, 0, stream>>>(stats1, 2 * HB, 0.0f);
    col_stats<<<512, 256, 0, stream>>>(y1, Nn, HB, stats1);
    bn_affine<<<1, HB, 0, stream>>>(stats1, g1, be1, alpha1, beta1, HB, 1.0f / Nn);
    // y2 = leaky( BN1(y1) @ W2 + b2 )                           (N,256)@(256,128)
    wmma_gemm<2, false, true, 256, 256, 128, 128>
        <<<dim3(Nn / 32, 1), 256, 0, stream>>>(y1, alpha1, beta1, W2, b2, y2);
    // BN2 stats -> final apply
    fill_kernel<<<1, 256, 0, stream>>>(stats2, 2 * D, 0.0f);
    col_stats<<<512, 256, 0, stream>>>(y2, Nn, D, stats2);
    bn_affine<<<1, D, 0, stream>>>(stats2, g2, be2, alpha2, beta2, D, 1.0f / Nn);
    bn_apply<<<(int)((ND / 4 + 255) / 256), 256, 0, stream>>>(
        (const float4*)y2, (const float4*)alpha2, (const float4*)beta2, (float4*)out, ND / 4);
}